// G_61856118997290
// MI455X (gfx1250) — compile-verified
//
#include <hip/hip_runtime.h>

// ---------------------------------------------------------------------------
// Dynamic per-pixel filtering (DUF-style) + softmax + pixel-shuffle, fused.
//
// out[b,c,4h+u/4,4w+u%4] = sum_p softmax_p(filt[b,p,u,h,w]) * patch[b,c,p,h,w]
//
// Memory-bound: filt = 719MB streamed once (~31us at 23.3 TB/s HBM).
// Per pixel: 16(u) x 16(n) x 343(taps) matmul on V_WMMA_F32_16X16X4_F32:
//   A[u,k] = exp(filt[tap k, u])     (f32, exp applied at consume time)
//   B[k,n] = patch (n<3) | 1.0 (n==3 => softmax denominator) | 0
// filt is streamed with GLOBAL_LOAD_ASYNC_TO_LDS_B128 (ASYNCcnt), double
// buffered one chunk ahead (s_wait_asynccnt 2). Last chunk (padded tap 343)
// is peeled so the hot loop carries no clamp/masking. B fragment is built
// with unconditional broadcast LDS reads + FMA masking (no EXEC divergence).
// ---------------------------------------------------------------------------

typedef __attribute__((ext_vector_type(2))) float v2f;
typedef __attribute__((ext_vector_type(8))) float v8f;

#define B_     2
#define T_     7
#define NHB    7
#define TAPS   343        // 7*7*7
#define UF2    16
#define H_     128
#define W_     128
#define HW     (H_ * W_)
#define STRIP  32         // pixels (consecutive w) per block
#define NCHUNK 86         // ceil(343/4); tap 343 padded (killed on B side)
#define XCOLS  40         // 32 + 6 halo + 2 pad
#define XTN    (3 * T_ * NHB * XCOLS)
#define EPAD   36         // pix stride in ecur: bank-conflict-free, 16B aligned
#define TAPSTR (UF2 * HW) // elements between consecutive taps (1 MB)

#if __has_builtin(__builtin_amdgcn_global_load_async_to_lds_b128)
#define USE_ASYNC_LDS 1
typedef int v4i __attribute__((vector_size(16)));
typedef __attribute__((address_space(3))) v4i v4i_lds;
#else
#define USE_ASYNC_LDS 0
#endif

#if __has_builtin(__builtin_amdgcn_s_wait_asynccnt)
#define WAIT_ASYNC(n) __builtin_amdgcn_s_wait_asynccnt(n)
#else
#define WAIT_ASYNC(n) asm volatile("s_wait_asynccnt %0" ::"i"(n) : "memory")
#endif

__global__ __launch_bounds__(256)
void dynfilter_wmma_kernel(const float* __restrict__ x,
                           const float* __restrict__ filt,
                           float* __restrict__ out)
{
    // x patch tile [c][t][kh][XCOLS], zero-padded halo:  3*7*7*40*4 = 23520 B
    __shared__ __align__(16) float xt[XTN];
    // raw filt chunk, double buffered: [buf][k][u][EPAD]  2*4*16*36*4 = 18432 B
    __shared__ __align__(16) float ecur[2][4 * UF2 * EPAD];
    // tap -> (t*7+kh)*40 + kw  (tbl[343] = 0 sentinel)            1376 B
    __shared__ int tbl[TAPS + 1];

    const int tid  = threadIdx.x;
    const int lane = tid & 31;
    const int wave = tid >> 5;

    // 1024 blocks: b (2) x h (128) x w-strip (4)
    const int strip = blockIdx.x;
    const int b  = strip >> 9;
    const int r  = strip & 511;
    const int h  = r >> 2;
    const int w0 = (r & 3) << 5;

    // ---- stage x tile into LDS (zero-padded borders) ----------------------
    for (int i = tid; i < XTN; i += 256) {
        int c    = i / (T_ * NHB * XCOLS);
        int rem  = i - c * (T_ * NHB * XCOLS);
        int t    = rem / (NHB * XCOLS);
        int rem2 = rem - t * (NHB * XCOLS);
        int kh   = rem2 / XCOLS;
        int col  = rem2 - kh * XCOLS;
        int row  = h + kh - 3;
        int cg   = w0 + col - 3;
        float v  = 0.0f;
        if ((unsigned)row < (unsigned)H_ && (unsigned)cg < (unsigned)W_)
            v = x[(((b * 3 + c) * T_ + t) * H_ + row) * W_ + cg];
        xt[i] = v;
    }
    // ---- tap decode table -------------------------------------------------
    for (int i = tid; i < TAPS + 1; i += 256) {
        int v = 0;
        if (i < TAPS) {
            int t  = i / 49;
            int rm = i - t * 49;
            int kh = rm / 7;
            int kw = rm - kh * 7;
            v = (t * 7 + kh) * XCOLS + kw;
        }
        tbl[i] = v;
    }

    // ---- per-thread filt-stream geometry (two b128 loads per chunk) -------
    // flat q = tid (and tid+256): pix4 = q&7, u = (q>>3)&15, k = q>>7
    const int uq   = (tid >> 3) & 15;
    const int pix4 = tid & 7;
    const int k0   = tid >> 7;                       // 0..1 ; second load k0+2
    const int lanepart = b * (TAPS * UF2 * HW) + uq * HW + h * W_ + w0 + 4 * pix4;
    const float* gb = filt + lanepart;               // + tap*TAPSTR per chunk
    const int eoff = (k0 * UF2 + uq) * EPAD + 4 * pix4;  // LDS dst (floats)

    // Fragment index helpers (wave32 WMMA layouts):
    //  A 16x4 f32 : lane L -> M = L&15 ; VGPR0/1 hold K = 2*(L>>4) + {0,1}
    //  B 4x16 f32 : lane L -> N = L&15 ; VGPR0/1 hold K = 2*(L>>4) + {0,1}
    //  C/D 16x16  : VGPR v, lane L -> M(u) = v + 8*(L>>4), N = L&15
    const int  un      = lane & 15;
    const int  grp     = lane >> 4;
    const bool ispatch = (un < 3);
    const int  xb      = (ispatch ? un : 0) * (T_ * NHB * XCOLS); // bcast clamp
    const float pm     = ispatch ? 1.0f : 0.0f;      // patch-column lane mask
    const float bden   = (un == 3) ? 1.0f : 0.0f;    // denominator column

    const v8f zero8 = {0.f,0.f,0.f,0.f,0.f,0.f,0.f,0.f};
    v8f acc[4] = {zero8, zero8, zero8, zero8};

    // ---- consume one staged chunk: 4 pixels per wave, 1 WMMA each ---------
    // B build: UNCONDITIONAL broadcast LDS reads, lane-masked via FMA so the
    // compiler cannot re-predicate the loads under EXEC:
    //   un<3 : p*1 + 0 = patch      un==3 : p*0 + 1 = 1      else : 0
    // `last` is a literal at each call site -> padded-tap masking folds away.
    auto consume = [&](const float* eb, int base, bool last) {
        const int tap0 = base + 2 * grp;
        const int a0   = xb + tbl[tap0];
        const int a1   = xb + tbl[tap0 + 1];          // tbl[343] = 0 sentinel
        const bool ok1   = !last || (tap0 + 1 < TAPS);
        const float pm1  = ok1 ? pm : 0.0f;           // also kills padded tap
        const float bden1 = ok1 ? bden : 0.0f;
        #pragma unroll
        for (int j = 0; j < 4; ++j) {
            const int pix = wave * 4 + j;
            v2f a, bb;
            a.x = __expf(eb[((2 * grp    ) * UF2 + un) * EPAD + pix]);
            a.y = __expf(eb[((2 * grp + 1) * UF2 + un) * EPAD + pix]);
            float p0 = xt[a0 + pix];                  // always finite, in-range
            float p1 = xt[a1 + pix];
            bb.x = __builtin_fmaf(p0, pm,  bden);
            bb.y = __builtin_fmaf(p1, pm1, bden1);
            acc[j] = __builtin_amdgcn_wmma_f32_16x16x4_f32(
                         false, a, false, bb, (short)0, acc[j], false, false);
        }
    };

#if USE_ASYNC_LDS
    // -- async DMA producer: global -> LDS, no VGPR round-trip --------------
    auto issue_fast = [&](int c) {                    // valid for c <= 84
        const float* s0 = gb + (size_t)(4 * c + k0) * TAPSTR;
        const float* s1 = s0 + (size_t)2 * TAPSTR;
        float* d0 = &ecur[c & 1][eoff];
        float* d1 = d0 + 2 * UF2 * EPAD;
        __builtin_amdgcn_global_load_async_to_lds_b128((v4i*)s0, (v4i_lds*)d0, 0, 0);
        __builtin_amdgcn_global_load_async_to_lds_b128((v4i*)s1, (v4i_lds*)d1, 0, 0);
    };
    auto issue_last = [&]() {                         // chunk 85: clamp tap 343
        const float* s0 = gb + (size_t)(340 + k0) * TAPSTR;
        const float* s1 = gb + (size_t)342 * TAPSTR;
        float* d0 = &ecur[(NCHUNK - 1) & 1][eoff];
        float* d1 = d0 + 2 * UF2 * EPAD;
        __builtin_amdgcn_global_load_async_to_lds_b128((v4i*)s0, (v4i_lds*)d0, 0, 0);
        __builtin_amdgcn_global_load_async_to_lds_b128((v4i*)s1, (v4i_lds*)d1, 0, 0);
    };

    __syncthreads();          // xt/tbl ready
    issue_fast(0);
    for (int c = 0; c < NCHUNK - 1; ++c) {           // chunks 0..84
        if (c + 2 < NCHUNK) issue_fast(c + 1);       // stay one chunk ahead
        else                issue_last();
        WAIT_ASYNC(2);                               // chunk c landed in LDS
        __syncthreads();
        consume(ecur[c & 1], 4 * c, false);
        __syncthreads();                             // done reading buf c
    }
    WAIT_ASYNC(0);                                   // peeled chunk 85
    __syncthreads();
    consume(ecur[(NCHUNK - 1) & 1], 4 * (NCHUNK - 1), true);
#else
    // -- fallback: vectorized b128 load -> regs -> b128 LDS store -----------
    __syncthreads();
    for (int c = 0; c < NCHUNK; ++c) {
        int t0 = 4 * c + k0;
        int t1 = t0 + 2;
        if (t1 > TAPS - 1) t1 = TAPS - 1;
        float4 v0 = *(const float4*)(gb + (size_t)t0 * TAPSTR);
        float4 v1 = *(const float4*)(gb + (size_t)t1 * TAPSTR);
        if (t0 + 4 < TAPS)     // prefetch next chunk (global_prefetch_b8)
            __builtin_prefetch(gb + (size_t)(t0 + 4) * TAPSTR, 0, 1);
        *(float4*)&ecur[0][eoff]                    = v0;
        *(float4*)&ecur[0][eoff + 2 * UF2 * EPAD]   = v1;
        __syncthreads();
        consume(ecur[0], 4 * c, c == NCHUNK - 1);
        __syncthreads();
    }
#endif

    // ---- epilogue: divide by denominator (column 3), pixel-shuffle scatter
    const int srcl = (lane & 16) | 3;   // lane holding D[u][3] in same group
    #pragma unroll
    for (int j = 0; j < 4; ++j) {
        const int pix = wave * 4 + j;
        const int w   = w0 + pix;
        #pragma unroll
        for (int v = 0; v < 8; ++v) {
            float num = acc[j][v];
            float den = __shfl(num, srcl, 32);
            if (un < 3) {                       // channels live in columns 0..2
                int u  = v + 8 * grp;
                int r1 = u >> 2, r2 = u & 3;
                long oi = ((long)((b * 3 + un) * (4 * H_) + (4 * h + r1)))
                          * (4 * W_) + (4 * w + r2);
                out[oi] = num / den;
            }
        }
    }
}

extern "C" void kernel_launch(void* const* d_in, const int* in_sizes, int n_in,
                              void* d_out, int out_size, void* d_ws, size_t ws_size,
                              hipStream_t stream) {
    const float* x    = (const float*)d_in[0];   // [2,3,7,128,128]
    const float* filt = (const float*)d_in[1];   // [2,343,16,128,128]
    float* out        = (float*)d_out;           // [2,3,512,512]
    (void)in_sizes; (void)n_in; (void)out_size; (void)d_ws; (void)ws_size;

    dim3 grid(B_ * H_ * (W_ / STRIP));   // 1024 blocks
    dynfilter_wmma_kernel<<<grid, 256, 0, stream>>>(x, filt, out);
}